// ResnetEncoderMatching_37142877176579
// MI455X (gfx1250) — compile-verified
//
#include <hip/hip_runtime.h>
#include <stdint.h>

#define BB 2
#define FF 2
#define CC 64
#define DD 96
#define HH 48
#define WW 160
#define HWSZ (HH * WW)
#define EPSV 1e-7f

// ---------------------------------------------------------------------------
// Kernel A: transpose [map][C][H*W] -> [map][H*W][C] (channel-last), LDS tiled
// ---------------------------------------------------------------------------
__global__ __launch_bounds__(256)
void transpose_cl_kernel(const float* __restrict__ src, float* __restrict__ dst) {
  __shared__ float tile[32][CC + 1];
  const int map = blockIdx.y;
  const int s0 = blockIdx.x * 32;
  const float* s = src + (size_t)map * CC * HWSZ;
  float* d = dst + (size_t)map * HWSZ * CC;
  const int tid = threadIdx.x;
#pragma unroll
  for (int i = 0; i < 8; ++i) {
    const int e = i * 256 + tid;
    const int c = e >> 5, sl = e & 31;           // coalesced read along spatial
    tile[sl][c] = s[(size_t)c * HWSZ + s0 + sl];
  }
  __syncthreads();
#pragma unroll
  for (int i = 0; i < 8; ++i) {
    const int e = i * 256 + tid;
    const int sl = e >> 6, c = e & 63;           // coalesced write along channel
    d[(size_t)(s0 + sl) * CC + c] = tile[sl][c];
  }
}

// ---------------------------------------------------------------------------
// Kernel B: main cost-volume kernel.
// Block = (b, y, 8 depth bins). 8 waves, wave = one depth bin, lane = 2 channels.
// Current-feature row (40KB, channel-last) staged via gfx1250 async-to-LDS.
// ---------------------------------------------------------------------------
__global__ __launch_bounds__(256)
void costvol_kernel(const float* __restrict__ curT,
                    const float* __restrict__ lukT,
                    const float* __restrict__ poses,
                    const float* __restrict__ Kmat,
                    const float* __restrict__ invKmat,
                    float* __restrict__ cost,
                    float* __restrict__ missing,
                    float* __restrict__ cv) {
  __shared__ float sh_cur[WW * CC];  // 40960 B: current row, channel-last

  const int bid = blockIdx.x;
  const int dchunk = bid % (DD / 8);
  const int y = (bid / (DD / 8)) % HH;
  const int b = bid / ((DD / 8) * HH);

  // ---- Stage current row into LDS with CDNA5 async loads (ASYNCcnt path) ----
  {
    const float* curRow = curT + ((size_t)b * HWSZ + (size_t)y * WW) * CC;
    const unsigned lds_base = (unsigned)(uintptr_t)(void*)&sh_cur[0];
    const unsigned long long gbase = (unsigned long long)(uintptr_t)curRow;
#pragma unroll
    for (int i = 0; i < 10; ++i) {  // 256 thr * 10 * 16B = 40960B
      const unsigned off = ((unsigned)threadIdx.x + (unsigned)i * 256u) * 16u;
      const unsigned lds_addr = lds_base + off;
      asm volatile("global_load_async_to_lds_b128 %0, %1, %2"
                   :: "v"(lds_addr), "v"(off), "s"(gbase) : "memory");
    }
    asm volatile("s_wait_asynccnt 0" ::: "memory");
  }
  __syncthreads();

  const int lane = threadIdx.x & 31;
  const int wave = threadIdx.x >> 5;
  const int d = dchunk * 8 + wave;
  const float dval = 0.1f + (float)d * ((20.0f - 0.1f) / 95.0f);  // linspace bin

  const float* iK = invKmat + (size_t)b * 16;
  const float ik00 = iK[0], ik01 = iK[1], ik02 = iK[2];
  const float ik10 = iK[4], ik11 = iK[5], ik12 = iK[6];
  const float ik20 = iK[8], ik21 = iK[9], ik22 = iK[10];

  // P = (K @ T)[:3] and pose-validity per frame (scalar loads, wave-uniform)
  float Pm[FF][12];
  float validf[FF];
  const float* Kb = Kmat + (size_t)b * 16;
#pragma unroll
  for (int f = 0; f < FF; ++f) {
    const float* T = poses + (size_t)(b * FF + f) * 16;
    float ts = 0.f;
#pragma unroll
    for (int e = 0; e < 16; ++e) ts += T[e];
    validf[f] = (ts != 0.f) ? 1.f : 0.f;
#pragma unroll
    for (int i = 0; i < 3; ++i)
#pragma unroll
      for (int j = 0; j < 4; ++j) {
        float s = 0.f;
#pragma unroll
        for (int k = 0; k < 4; ++k) s += Kb[i * 4 + k] * T[k * 4 + j];
        Pm[f][i * 4 + j] = s;
      }
  }

  const float fy = (float)y;
  const bool yin = (y >= 2) && (y < HH - 2);

  float res[WW / 32];
#pragma unroll
  for (int xc = 0; xc < WW / 32; ++xc) {
    float mine = 0.f;
    for (int xi = 0; xi < 32; ++xi) {
      const int x = xc * 32 + xi;
      const float fx = (float)x;
      // ray = invK[:3,:3] @ [x,y,1];  pts = d*ray (homog 1 handled in cp)
      const float r0 = ik00 * fx + ik01 * fy + ik02;
      const float r1 = ik10 * fx + ik11 * fy + ik12;
      const float r2 = ik20 * fx + ik21 * fy + ik22;
      const float q0 = dval * r0, q1 = dval * r1, q2 = dval * r2;
      const bool interior = yin && (x >= 2) && (x < WW - 2);
      const float2 curv = *(const float2*)&sh_cur[x * CC + lane * 2];  // ds_load_b64
      float dsum[FF];
#pragma unroll
      for (int f = 0; f < FF; ++f) {
        const float cp0 = Pm[f][0] * q0 + Pm[f][1] * q1 + Pm[f][2] * q2 + Pm[f][3];
        const float cp1 = Pm[f][4] * q0 + Pm[f][5] * q1 + Pm[f][6] * q2 + Pm[f][7];
        const float cp2 = Pm[f][8] * q0 + Pm[f][9] * q1 + Pm[f][10] * q2 + Pm[f][11];
        const float z = cp2 + EPSV;
        const float px = cp0 / z;
        const float py = cp1 / z;
        float df = 0.f;
        // wave-uniform edge mask -> uniform branch skips all gathers when masked
        const bool edge = interior && (px >= 2.0f) && (px <= (float)(WW - 2)) &&
                          (py >= 2.0f) && (py <= (float)(HH - 2));
        if (edge) {
          const float x0f = floorf(px), y0f = floorf(py);
          const int x0 = (int)x0f, y0 = (int)y0f;
          const float wx = px - x0f, wy = py - y0f;
          const float w00 = (1.f - wx) * (1.f - wy);
          const float w10 = wx * (1.f - wy);
          const float w01 = (1.f - wx) * wy;
          const float w11 = wx * wy;
          // channel-last corners: wave-uniform base, coalesced 256B per corner
          const float* bp = lukT +
              ((size_t)(b * FF + f) * HWSZ + (size_t)y0 * WW + x0) * CC + lane * 2;
          const float2 c00 = *(const float2*)(bp);
          const float2 c10 = *(const float2*)(bp + CC);
          const float2 c01 = *(const float2*)(bp + (size_t)WW * CC);
          const float2 c11 = *(const float2*)(bp + (size_t)WW * CC + CC);
          const float wa = w00 * c00.x + w10 * c10.x + w01 * c01.x + w11 * c11.x;
          const float wb = w00 * c00.y + w10 * c10.y + w01 * c01.y + w11 * c11.y;
          df = fabsf(wa - curv.x) + fabsf(wb - curv.y);
        }
        // wave32 tree reduction over 32 lanes (64 channels)
#pragma unroll
        for (int m = 16; m >= 1; m >>= 1) df += __shfl_xor(df, m, 32);
        dsum[f] = df * (1.0f / 64.0f) * validf[f];
      }
      const float cnt = (dsum[0] > 0.f ? 1.f : 0.f) + (dsum[1] > 0.f ? 1.f : 0.f);
      const float cval = (dsum[0] + dsum[1]) / (cnt + EPSV);
      mine = (xi == lane) ? cval : mine;  // stash per-x result in its lane
    }
    res[xc] = mine;
  }

  // coalesced stores: 32 consecutive x per chunk
  const size_t obase = (((size_t)b * DD + d) * HH + y) * WW;
#pragma unroll
  for (int xc = 0; xc < WW / 32; ++xc) {
    const size_t idx = obase + (size_t)xc * 32 + lane;
    const float cval = res[xc];
    cost[idx] = cval;
    cv[idx] = cval;                              // cv_feature = raw cost
    missing[idx] = (cval == 0.f) ? 1.f : 0.f;    // missing indicator
  }
}

// ---------------------------------------------------------------------------
// Kernel C: set_missing_to_max — per (b,pixel) max over D, replace zeros
// ---------------------------------------------------------------------------
__global__ __launch_bounds__(256)
void maxfix_kernel(float* __restrict__ cost) {
  const int t = blockIdx.x * blockDim.x + threadIdx.x;
  if (t >= BB * HWSZ) return;
  const int b = t / HWSZ;
  const int hw = t % HWSZ;
  const size_t base = (size_t)b * DD * HWSZ + hw;
  float m = 0.f;  // cost >= 0 always
  for (int dd = 0; dd < DD; ++dd) m = fmaxf(m, cost[base + (size_t)dd * HWSZ]);
  for (int dd = 0; dd < DD; ++dd) {
    const size_t idx = base + (size_t)dd * HWSZ;
    if (cost[idx] == 0.f) cost[idx] = m;
  }
}

// ---------------------------------------------------------------------------
extern "C" void kernel_launch(void* const* d_in, const int* in_sizes, int n_in,
                              void* d_out, int out_size, void* d_ws, size_t ws_size,
                              hipStream_t stream) {
  (void)in_sizes; (void)n_in; (void)out_size; (void)ws_size;
  const float* cur   = (const float*)d_in[0];  // [B,C,H,W]
  const float* luk   = (const float*)d_in[1];  // [B,F,C,H,W]
  const float* poses = (const float*)d_in[2];  // [B,F,4,4]
  const float* Kmat  = (const float*)d_in[3];  // [B,4,4]
  const float* invK  = (const float*)d_in[4];  // [B,4,4]

  float* out = (float*)d_out;
  float* cost    = out;
  float* missing = out + (size_t)BB * DD * HWSZ;
  float* cv      = out + 2 * (size_t)BB * DD * HWSZ;

  float* ws = (float*)d_ws;
  float* curT = ws;                                  // [B][H*W][C]
  float* lukT = ws + (size_t)BB * HWSZ * CC;         // [B*F][H*W][C]

  transpose_cl_kernel<<<dim3(HWSZ / 32, BB), 256, 0, stream>>>(cur, curT);
  transpose_cl_kernel<<<dim3(HWSZ / 32, BB * FF), 256, 0, stream>>>(luk, lukT);
  costvol_kernel<<<BB * HH * (DD / 8), 256, 0, stream>>>(
      curT, lukT, poses, Kmat, invK, cost, missing, cv);
  maxfix_kernel<<<(BB * HWSZ + 255) / 256, 256, 0, stream>>>(cost);
}